// Pre_VQ_29600914604554
// MI455X (gfx1250) — compile-verified
//
#include <hip/hip_runtime.h>

#define GRP 8
#define CG 128
#define EG 64
#define TT 64            // T-tile width per compute step
#define TILES_PER_BLK 4  // persistent: block handles 4 consecutive T-tiles
#define TDIM 1024
#define CDIM 1024
#define EDIM 512
#define NELEM (CG * TDIM)

#define W1S 132          // w1 LDS row stride (132 % 64 == 4 -> conflict-free A reads)
#define XSS 80           // x LDS row stride (half-waves hit disjoint bank groups)
#define W2SS 132
#define HTS 132          // transposed hn tile stride: hnT[t][k], contiguous-k B frags

typedef __attribute__((ext_vector_type(2))) float v2f;
typedef __attribute__((ext_vector_type(8))) float v8f;

// sizes in floats
#define SZ_W1  (CG * W1S)          // 16896
#define SZ_XS  (CG * XSS)          // 10240 (also holds hnT: 64*132 = 8448 <= 10240)
#define SZ_W2  (EG * W2SS)         // 8448

#define WAIT_ASYNC(n) asm volatile("s_wait_asynccnt %0" :: "i"(n) : "memory")

__device__ __forceinline__ uint32_t lds_addr_of(const float* p) {
  // generic shared pointer = SHARED_BASE aperture | 32-bit LDS offset (ISA 10.2)
  return (uint32_t)(uintptr_t)p;
}

__device__ __forceinline__ void async_b128(uint32_t lds, uint32_t voff, const void* sbase) {
  // GLOBAL_LOAD_ASYNC_TO_LDS_B128, GVS mode: mem = SADDR + VADDR(i32); LDS dest = VDST vgpr
  asm volatile("global_load_async_to_lds_b128 %0, %1, %2"
               :: "v"(lds), "v"(voff), "s"(sbase) : "memory");
}

// ---------------- async staging: global -> LDS DMA, 16B chunks ----------------

__device__ __forceinline__ void stage_w1_async(uint32_t lds0, const float* w1g, int tid) {
#pragma unroll
  for (int i = 0; i < 16; ++i) {             // 16 wave-instructions
    int fi = tid + i * 256;
    uint32_t va = (uint32_t)(fi >> 5) * (CG * 4u) + (uint32_t)(fi & 31) * 16u;
    uint32_t la = lds0 + (uint32_t)(fi >> 5) * (W1S * 4u) + (uint32_t)(fi & 31) * 16u;
    async_b128(la, va, w1g);
  }
}

__device__ __forceinline__ void stage_w2_async(uint32_t lds0, const float* w2g, int tid) {
#pragma unroll
  for (int i = 0; i < 8; ++i) {              // 8 wave-instructions
    int fi = tid + i * 256;
    uint32_t va = (uint32_t)(fi >> 5) * (CG * 4u) + (uint32_t)(fi & 31) * 16u;
    uint32_t la = lds0 + (uint32_t)(fi >> 5) * (W2SS * 4u) + (uint32_t)(fi & 31) * 16u;
    async_b128(la, va, w2g);
  }
}

__device__ __forceinline__ void stage_x_async(uint32_t lds0, const float* slab, int tid) {
#pragma unroll
  for (int i = 0; i < 8; ++i) {              // 8 wave-instructions
    int fi = tid + i * 256;
    uint32_t va = (uint32_t)(fi >> 4) * (TDIM * 4u) + (uint32_t)(fi & 15) * 16u;
    uint32_t la = lds0 + (uint32_t)(fi >> 4) * (XSS * 4u) + (uint32_t)(fi & 15) * 16u;
    async_b128(la, va, slab);
  }
}

// ---------------- conv1 WMMA core: wave computes h[mbase:+16, 0:64] ----------------

__device__ __forceinline__ void conv1_compute(const float* w1s, const float* xs,
                                              int mbase, int lane, v8f acc[4]) {
  const int r  = lane & 15;
  const int kb = (lane >> 4) << 1;
  const float* w1row = w1s + (size_t)(mbase + r) * W1S;
#pragma unroll 8
  for (int k = 0; k < CG; k += 4) {
    v2f a = *(const v2f*)&w1row[k + kb];
#pragma unroll
    for (int n = 0; n < 4; ++n) {
      v2f bb;
      bb.x = xs[(k + kb)     * XSS + n * 16 + r];
      bb.y = xs[(k + kb + 1) * XSS + n * 16 + r];
      acc[n] = __builtin_amdgcn_wmma_f32_16x16x4_f32(
          false, a, false, bb, (short)0, acc[n], false, false);
    }
  }
}

// ---------------- kernel 1: conv1 + GroupNorm partial stats (double-buffered) ----------------

__global__ __launch_bounds__(256) void conv1_stats_kernel(
    const float* __restrict__ x, const float* __restrict__ w1,
    const float* __restrict__ b1, float* __restrict__ sums,
    float* __restrict__ sumsqs) {
  extern __shared__ float sm[];
  float* w1s = sm;                       // SZ_W1
  float* xs0 = sm + SZ_W1;               // SZ_XS
  float* xs1 = xs0 + SZ_XS;              // SZ_XS
  float* red = xs1 + SZ_XS;              // 512

  const int tid = threadIdx.x;
  const int q = blockIdx.x, bg = blockIdx.y;
  const int b = bg >> 3, g = bg & 7;
  const int tbase0 = q * (TT * TILES_PER_BLK);
  const float* slab = x + ((size_t)b * CDIM + (size_t)g * CG) * TDIM;

  const uint32_t lw1 = lds_addr_of(w1s);
  const uint32_t lx[2] = {lds_addr_of(xs0), lds_addr_of(xs1)};
  float* xsp[2] = {xs0, xs1};

  stage_w1_async(lw1, w1 + (size_t)g * CG * CG, tid);          // 16/wave
  stage_x_async(lx[0], slab + tbase0, tid);                    // 8/wave

  const int wid = tid >> 5, lane = tid & 31;
  const int mbase = wid << 4;
  const int rb = mbase + ((lane >> 4) << 3);
  float4 ba = *(const float4*)&b1[g * CG + rb];
  float4 bbv = *(const float4*)&b1[g * CG + rb + 4];
  float bias[8] = {ba.x, ba.y, ba.z, ba.w, bbv.x, bbv.y, bbv.z, bbv.w};

  float s = 0.f, s2 = 0.f;

#pragma unroll
  for (int t = 0; t < TILES_PER_BLK; ++t) {
    if (t + 1 < TILES_PER_BLK)
      stage_x_async(lx[(t + 1) & 1], slab + tbase0 + (t + 1) * TT, tid);
    if (t + 1 < TILES_PER_BLK) { WAIT_ASYNC(8); } else { WAIT_ASYNC(0); }
    __syncthreads();                               // current buffer fully in LDS

    v8f acc[4] = {};
    conv1_compute(w1s, xsp[t & 1], mbase, lane, acc);
#pragma unroll
    for (int n = 0; n < 4; ++n)
#pragma unroll
      for (int v = 0; v < 8; ++v) {
        float h = acc[n][v] + bias[v];
        s += h;
        s2 += h * h;
      }
    __syncthreads();                               // reads done before next DMA lands
  }

  red[tid] = s;
  red[256 + tid] = s2;
  __syncthreads();
  for (int w = 128; w > 0; w >>= 1) {
    if (tid < w) {
      red[tid] += red[tid + w];
      red[256 + tid] += red[256 + tid + w];
    }
    __syncthreads();
  }
  if (tid == 0) {
    atomicAdd(&sums[bg], red[0]);
    atomicAdd(&sumsqs[bg], red[256]);
  }
}

// ---------------- kernel 2: finalize mean / rstd ----------------

__global__ void finalize_stats_kernel(const float* __restrict__ sums,
                                      const float* __restrict__ sumsqs,
                                      float* __restrict__ meanp,
                                      float* __restrict__ rstdp) {
  int i = threadIdx.x;
  if (i < 256) {
    const float inv = 1.0f / (float)NELEM;
    float m = sums[i] * inv;
    float v = sumsqs[i] * inv - m * m;
    meanp[i] = m;
    rstdp[i] = rsqrtf(v + 1e-5f);
  }
}

__global__ void zero_ws_kernel(float* __restrict__ p, int n) {
  int i = blockIdx.x * blockDim.x + threadIdx.x;
  if (i < n) p[i] = 0.f;
}

// ---------------- kernel 3: recompute conv1, normalize+ReLU, conv2 (persistent) ----------------

__global__ __launch_bounds__(256) void fused_norm_conv2_kernel(
    const float* __restrict__ x, const float* __restrict__ w1,
    const float* __restrict__ b1, const float* __restrict__ gamma,
    const float* __restrict__ beta, const float* __restrict__ w2,
    const float* __restrict__ b2, const float* __restrict__ meanp,
    const float* __restrict__ rstdp, float* __restrict__ out) {
  extern __shared__ float sm[];
  float* w1s = sm;                       // SZ_W1
  float* w2s = sm + SZ_W1;               // SZ_W2
  float* xs  = w2s + SZ_W2;              // SZ_XS: x tile, then reused as hnT[64][HTS]
  float* hnT = xs;

  const int tid = threadIdx.x;
  const int q = blockIdx.x, bg = blockIdx.y;
  const int b = bg >> 3, g = bg & 7;
  const int tbase0 = q * (TT * TILES_PER_BLK);
  const float* slab = x + ((size_t)b * CDIM + (size_t)g * CG) * TDIM;

  const uint32_t lw1 = lds_addr_of(w1s);
  const uint32_t lw2 = lds_addr_of(w2s);
  const uint32_t lxs = lds_addr_of(xs);

  stage_w1_async(lw1, w1 + (size_t)g * CG * CG, tid);
  stage_w2_async(lw2, w2 + (size_t)g * EG * CG, tid);

  const int wid = tid >> 5, lane = tid & 31;
  const int mbase = wid << 4;
  const int col = lane & 15;
  const float mean = meanp[bg], rstd = rstdp[bg];

  const int rb = mbase + ((lane >> 4) << 3);
  const int cbase = g * CG + rb;
  float4 b1a = *(const float4*)&b1[cbase],    b1b = *(const float4*)&b1[cbase + 4];
  float4 ga  = *(const float4*)&gamma[cbase], gb  = *(const float4*)&gamma[cbase + 4];
  float4 ea  = *(const float4*)&beta[cbase],  eb  = *(const float4*)&beta[cbase + 4];
  float bias[8] = {b1a.x, b1a.y, b1a.z, b1a.w, b1b.x, b1b.y, b1b.z, b1b.w};
  float gam[8]  = {ga.x, ga.y, ga.z, ga.w, gb.x, gb.y, gb.z, gb.w};
  float bet[8]  = {ea.x, ea.y, ea.z, ea.w, eb.x, eb.y, eb.z, eb.w};

  const int wm = wid & 3, wn = wid >> 2;
  const int erb = wm * 16 + ((lane >> 4) << 3);
  float4 c0 = *(const float4*)&b2[g * EG + erb];
  float4 c1 = *(const float4*)&b2[g * EG + erb + 4];
  float b2v[8] = {c0.x, c0.y, c0.z, c0.w, c1.x, c1.y, c1.z, c1.w};

#pragma unroll
  for (int t = 0; t < TILES_PER_BLK; ++t) {
    const int tbase = tbase0 + t * TT;
    stage_x_async(lxs, slab + tbase, tid);
    WAIT_ASYNC(0);                        // first iter also covers w1/w2 DMA
    __syncthreads();

    v8f acc[4] = {};
    conv1_compute(w1s, xs, mbase, lane, acc);
    __syncthreads();                      // all conv1 reads of xs done

    // normalize + ReLU; write TRANSPOSED hn tile: hnT[t_local][k_channel]
    // (free relayout -> conv2 B fragments become contiguous 64-bit LDS loads)
#pragma unroll
    for (int v = 0; v < 8; ++v) {
      const int row = rb + v;             // k-channel index
      const float sc  = gam[v] * rstd;
      const float off = bet[v] - mean * sc;
#pragma unroll
      for (int n = 0; n < 4; ++n) {
        float h = acc[n][v] + bias[v];
        hnT[(n * 16 + col) * HTS + row] = fmaxf(h * sc + off, 0.f);
      }
    }
    __syncthreads();

    // conv2: 64x64 tile; 4x2 wave grid, 2 N-tiles per wave
    const int r  = lane & 15;
    const int kb = (lane >> 4) << 1;
    const float* w2row = w2s + (size_t)(wm * 16 + r) * W2SS;
    const float* hrow0 = hnT + (size_t)(wn * 32 + r) * HTS;        // j = 0
    const float* hrow1 = hnT + (size_t)(wn * 32 + 16 + r) * HTS;   // j = 1
    v8f acc2[2] = {};
#pragma unroll 8
    for (int k = 0; k < CG; k += 4) {
      v2f a  = *(const v2f*)&w2row[k + kb];
      v2f b0 = *(const v2f*)&hrow0[k + kb];
      v2f b1f = *(const v2f*)&hrow1[k + kb];
      acc2[0] = __builtin_amdgcn_wmma_f32_16x16x4_f32(
          false, a, false, b0, (short)0, acc2[0], false, false);
      acc2[1] = __builtin_amdgcn_wmma_f32_16x16x4_f32(
          false, a, false, b1f, (short)0, acc2[1], false, false);
    }

    float* ybase = out + (size_t)b * EDIM * TDIM + (size_t)(g * EG) * TDIM + tbase;
#pragma unroll
    for (int v = 0; v < 8; ++v) {
      const int e = erb + v;
#pragma unroll
      for (int j = 0; j < 2; ++j) {
        ybase[(size_t)e * TDIM + wn * 32 + j * 16 + col] = acc2[j][v] + b2v[v];
      }
    }
    __syncthreads();                      // hnT reads done before next tile's DMA
  }
}

// ---------------- host launcher ----------------

extern "C" void kernel_launch(void* const* d_in, const int* in_sizes, int n_in,
                              void* d_out, int out_size, void* d_ws, size_t ws_size,
                              hipStream_t stream) {
  const float* x     = (const float*)d_in[0];
  const float* w1    = (const float*)d_in[1];
  const float* b1    = (const float*)d_in[2];
  const float* gamma = (const float*)d_in[3];
  const float* beta  = (const float*)d_in[4];
  const float* w2    = (const float*)d_in[5];
  const float* b2    = (const float*)d_in[6];
  float* out = (float*)d_out;

  float* sums   = (float*)d_ws;        // 256
  float* sumsqs = sums + 256;          // 256
  float* meanp  = sums + 512;          // 256
  float* rstdp  = sums + 768;          // 256

  zero_ws_kernel<<<4, 256, 0, stream>>>(sums, 1024);

  dim3 grid(TDIM / (TT * TILES_PER_BLK), 32 * GRP);   // 4 x 256 persistent blocks
  size_t shm1 = (size_t)(SZ_W1 + 2 * SZ_XS + 512) * sizeof(float);
  conv1_stats_kernel<<<grid, 256, shm1, stream>>>(x, w1, b1, sums, sumsqs);

  finalize_stats_kernel<<<1, 256, 0, stream>>>(sums, sumsqs, meanp, rstdp);

  size_t shm3 = (size_t)(SZ_W1 + SZ_W2 + SZ_XS) * sizeof(float);
  fused_norm_conv2_kernel<<<grid, 256, shm3, stream>>>(
      x, w1, b1, gamma, beta, w2, b2, meanp, rstdp, out);
}